// LiquidRON_15513421873384
// MI455X (gfx1250) — compile-verified
//
#include <hip/hip_runtime.h>

typedef __attribute__((ext_vector_type(2))) float v2f;
typedef __attribute__((ext_vector_type(8))) float v8f;

#define NN 1000   // neurons (M and K of the GEMM)
#define BB 64     // batch   (N of the GEMM)
#define TT 1000   // timesteps
#define THRESH 30.0f

#define NBLK 32
#define NTHR 256
#define NT   (NBLK * NTHR)          // 8192 threads, 256 waves
#define TILES_B (BB / 16)           // 4
#define TILES_M ((NN + 15) / 16)    // 63
#define NTILES  (TILES_B * TILES_M) // 252

// ---------------------------------------------------------------------------
// ws layout (floats): v[NN*BB] | u[NN*BB] | sf[NN*BB] | zrow[NN] | sync[2]
// v,u,sf in [n][b] order; zrow = zeroed dummy S-row for M-tail padding;
// sync = {arrive counter, generation} for the grid-wide barrier.
// ---------------------------------------------------------------------------

__global__ void init_ws(float* __restrict__ vws, float* __restrict__ uws,
                        float* __restrict__ zrow, unsigned* __restrict__ sync) {
  int i = blockIdx.x * blockDim.x + threadIdx.x;
  if (i < NN * BB) { vws[i] = 0.0f; uws[i] = 0.0f; }
  if (i < NN) zrow[i] = 0.0f;
  if (i < 2) sync[i] = 0u;
}

// Grid-wide two-phase barrier (all NBLK blocks are co-resident).
__device__ __forceinline__ void grid_sync(unsigned* cnt, unsigned* gen) {
  __threadfence();          // make this thread's global stores visible
  __syncthreads();
  if (threadIdx.x == 0) {
    volatile unsigned* vgen = (volatile unsigned*)gen;
    unsigned g = *vgen;
    if (atomicAdd(cnt, 1u) == (unsigned)(NBLK - 1)) {
      *cnt = 0u;            // reset for next barrier
      __threadfence();
      atomicAdd(gen, 1u);   // release
    } else {
      while (*vgen == g) __builtin_amdgcn_s_sleep(2);
    }
  }
  __syncthreads();
}

__global__ __launch_bounds__(NTHR)
void liquid_persistent(const float* __restrict__ data,   // [T,B,N]
                       const float* __restrict__ Uv,
                       const float* __restrict__ S,      // [N,N]
                       const float* __restrict__ aarr,
                       const float* __restrict__ barr,
                       const float* __restrict__ carr,
                       const float* __restrict__ darr,
                       float* __restrict__ vws, float* __restrict__ uws,
                       float* __restrict__ sfb,
                       const float* __restrict__ zrow,
                       unsigned* __restrict__ cnt, unsigned* __restrict__ gen,
                       float* __restrict__ states_out,   // [T,B,N]
                       float* __restrict__ v_out,        // [B,N]
                       float* __restrict__ u_out,        // [B,N]
                       float* __restrict__ spikes_out) { // [T,B,N]
  const int gtid = blockIdx.x * NTHR + threadIdx.x;
  const int lane = threadIdx.x & 31;
  const int wv   = gtid >> 5;            // global wave id (uniform per wave)
  const int lh   = lane & 15;
  const int hi   = lane >> 4;

  // Tile assignment for phase B (one 16x16 output tile per wave).
  const bool have_tile = (wv < NTILES);
  const int tm   = wv / TILES_B;
  const int tb   = wv - tm * TILES_B;
  const int row0 = tm * 16;
  const int col0 = tb * 16;
  const int arow = row0 + lh;
  const float* __restrict__ Sr =
      (have_tile && arow < NN) ? (S + (size_t)arow * NN) : zrow;
  const float* __restrict__ Bp = sfb + col0 + lh;
  const int col = col0 + lh;

  for (int t = 0; t < TT; ++t) {
    // ---- Phase A: spike detect + reset; stage sf in [k][b] layout ----
    {
      float* __restrict__ spk = spikes_out + (size_t)t * BB * NN;
      for (int i = gtid; i < NN * BB; i += NT) {
        int n = i / BB;
        int b = i - n * BB;
        float v = vws[i];
        float u = uws[i];
        float s = (v >= THRESH) ? 1.0f : 0.0f;
        if (s != 0.0f) { v = carr[n]; u = u + darr[n]; }
        vws[i] = v;
        uws[i] = u;
        sfb[i] = s;                 // B-operand row k=n, col b
        spk[(size_t)b * NN + n] = s;
      }
    }
    grid_sync(cnt, gen);

    // ---- Phase B: D = S * sf via V_WMMA_F32_16X16X4_F32 + state update ----
    if (have_tile) {                // wave-uniform: EXEC stays all-ones
      const float* __restrict__ x_t = data + (size_t)t * BB * NN;
      v8f acc = {};
      for (int k = 0; k < NN; k += 4) {
        int kk = k + 2 * hi;
        v2f aop = *(const v2f*)(Sr + kk);       // b64, 8-byte aligned
        v2f bop;
        bop.x = Bp[(size_t)kk * BB];
        bop.y = Bp[(size_t)(kk + 1) * BB];
        acc = __builtin_amdgcn_wmma_f32_16x16x4_f32(
            false, aop, false, bop, (short)0, acc, false, false);
      }
      float* __restrict__ st = states_out + (size_t)t * BB * NN;
#pragma unroll
      for (int j = 0; j < 8; ++j) {
        int row = row0 + j + 8 * hi;
        if (row >= NN) continue;
        float I = x_t[(size_t)col * NN + row] * Uv[row] + acc[j];
        size_t idx = (size_t)row * BB + col;
        float v = vws[idx];
        float u = uws[idx];
        v = v + 0.5f * (0.04f * v * v + 5.0f * v + 140.0f - u + I);
        u = u + aarr[row] * (barr[row] * v - u);
        vws[idx] = v;
        uws[idx] = u;
        st[(size_t)col * NN + row] = (v >= THRESH) ? 1.0f : 0.0f;
      }
    }
    grid_sync(cnt, gen);
  }

  // ---- Final state copy: [n][b] ws -> [b][n] outputs ----
  for (int i = gtid; i < NN * BB; i += NT) {
    int n = i / BB;
    int b = i - n * BB;
    v_out[(size_t)b * NN + n] = vws[i];
    u_out[(size_t)b * NN + n] = uws[i];
  }
}

extern "C" void kernel_launch(void* const* d_in, const int* in_sizes, int n_in,
                              void* d_out, int out_size, void* d_ws, size_t ws_size,
                              hipStream_t stream) {
  // setup_inputs order: data, U, S, a, b, c, d
  const float* data = (const float*)d_in[0];
  const float* Uv   = (const float*)d_in[1];
  const float* S    = (const float*)d_in[2];
  const float* aarr = (const float*)d_in[3];
  const float* barr = (const float*)d_in[4];
  const float* carr = (const float*)d_in[5];
  const float* darr = (const float*)d_in[6];

  float* out        = (float*)d_out;
  float* states_out = out;                                   // T*B*N
  float* v_out      = states_out + (size_t)TT * BB * NN;     // B*N
  float* u_out      = v_out + (size_t)BB * NN;               // B*N
  float* spikes_out = u_out + (size_t)BB * NN;               // T*B*N

  float*    vws  = (float*)d_ws;
  float*    uws  = vws + (size_t)NN * BB;
  float*    sfb  = uws + (size_t)NN * BB;
  float*    zrow = sfb + (size_t)NN * BB;
  unsigned* sync = (unsigned*)(zrow + NN);

  init_ws<<<(NN * BB + NTHR - 1) / NTHR, NTHR, 0, stream>>>(vws, uws, zrow, sync);

  liquid_persistent<<<NBLK, NTHR, 0, stream>>>(
      data, Uv, S, aarr, barr, carr, darr,
      vws, uws, sfb, zrow, sync, sync + 1,
      states_out, v_out, u_out, spikes_out);
}